// BotSpot_28020366639117
// MI455X (gfx1250) — compile-verified
//
#include <hip/hip_runtime.h>
#include <hip/hip_bf16.h>
#include <stdint.h>

typedef float v2f __attribute__((ext_vector_type(2)));
typedef float v4f __attribute__((ext_vector_type(4)));
typedef float v8f __attribute__((ext_vector_type(8)));

#define ND 200000
#define NC 1000
#define BSZ 65536

// ---------------- ws layout (float offsets) ----------------
// dev_emb [ND][128]          @ 0                (25,600,000)
// partials (aliased later by z/z1/z2):
//   psum [256][500*128]      @ 25,600,000       (16,384,000)
//   z    [B][112]            @ 25,600,000       (7,340,032)   (after agg done)
//   z1   [B][64]             @ 32,940,032       (4,194,304)
//   z2   [B][32]             @ 37,134,336       (2,097,152)
// pdeg [256][500]            @ 41,984,000       (128,000)
// nbr_mean [NC][128]         @ 42,112,000       (128,000)
// fus_src [NC][84]           @ 42,240,000       (84,000)
// h [B][80]                  @ 42,324,000       (5,242,880)   -> total 47,566,880 floats (~191 MB)
#define OFF_DEVEMB 0ull
#define OFF_PSUM   25600000ull
#define OFF_Z      25600000ull
#define OFF_Z1     32940032ull
#define OFF_Z2     37134336ull
#define OFF_PDEG   41984000ull
#define OFF_NBR    42112000ull
#define OFF_FUS    42240000ull
#define OFF_H      42324000ull

// ---------------- build dev_emb [d][0..112], zero-pad to 128 ----------------
__global__ __launch_bounds__(256) void k_build(const float* __restrict__ dnum,
    const int* __restrict__ dcat,
    const float* __restrict__ e0, const float* __restrict__ e1,
    const float* __restrict__ e2, const float* __restrict__ e3,
    const float* __restrict__ e4, const float* __restrict__ e5,
    const float* __restrict__ e6, float* __restrict__ dev_emb)
{
    int d = blockIdx.x * 2 + (threadIdx.x >> 7);
    int t = threadIdx.x & 127;
    float val = 0.f;
    if (t == 0) {
        val = dnum[d];
    } else if (t < 113) {
        int i = (t - 1) >> 4, j = (t - 1) & 15;
        const float* tabs[7] = {e0, e1, e2, e3, e4, e5, e6};
        int cat = dcat[d * 7 + i];
        val = tabs[i][cat * 16 + j];
    }
    dev_emb[(size_t)d * 128 + t] = val;
}

// ---------------- sparse aggregation: LDS-accumulated column sums ----------------
// grid = 256 (2 channel-halves x 128 device stripes), 256 threads, ~262KB dyn LDS.
__global__ __launch_bounds__(256) void k_agg(const uint32_t* __restrict__ adj32,
    const float* __restrict__ dev_emb,
    float* __restrict__ psum, float* __restrict__ pdeg)
{
    extern __shared__ float smem[];
    float* acc = smem;                           // 500*128 = 64000
    float* dg  = smem + 64000;                   // 500
    uint32_t* q = (uint32_t*)(dg + 500);         // 1024 queue + [1024]=counter
    const int tid  = threadIdx.x;
    const int half = blockIdx.x >> 7;
    const int wg   = blockIdx.x & 127;
    for (int i = tid; i < 64500; i += 256) smem[i] = 0.f;
    if (tid == 0) q[1024] = 0u;
    __syncthreads();

    const int DPW = (ND + 127) / 128;            // 1563
    const int dstart = wg * DPW;
    const int dend   = min(dstart + DPW, ND);
    const int wave = tid >> 5, l = tid & 31;

    for (int d0 = dstart; d0 < dend; d0 += 8) {
        // scan adj rows (this half's 125 dwords per row), enqueue nonzero bytes
        for (int it = 0; it < 4; ++it) {
            int ix = it * 256 + tid;
            if (ix < 1000) {
                int dd = ix / 125, w = ix - dd * 125;
                int d = d0 + dd;
                if (d < dend) {
                    uint32_t word = adj32[(size_t)d * 250 + half * 125 + w];
                    if (word) {
                        #pragma unroll
                        for (int b = 0; b < 4; ++b) {
                            if ((word >> (8 * b)) & 0xFFu) {
                                int cl = w * 4 + b;
                                uint32_t pos = atomicAdd(&q[1024], 1u);
                                if (pos < 1024u) q[pos] = ((uint32_t)d << 9) | (uint32_t)cl;
                                atomicAdd(&dg[cl], 1.f);
                            }
                        }
                    }
                }
            }
        }
        __syncthreads();
        uint32_t cnt = min(q[1024], 1024u);
        // wave-per-edge: 32 lanes x float4 covers the whole padded 128-col row
        for (uint32_t e = (uint32_t)wave; e < cnt; e += 8u) {
            uint32_t qe = q[e];
            int cl = (int)(qe & 511u);
            int d  = (int)(qe >> 9);
            v4f x = *reinterpret_cast<const v4f*>(dev_emb + (size_t)d * 128 + l * 4);
            float* ap = acc + cl * 128 + l * 4;
            atomicAdd(ap + 0, x[0]); atomicAdd(ap + 1, x[1]);
            atomicAdd(ap + 2, x[2]); atomicAdd(ap + 3, x[3]);
        }
        __syncthreads();
        if (tid == 0) q[1024] = 0u;
        __syncthreads();
    }
    // atomic-free flush to per-WG partials
    size_t pbase = (size_t)blockIdx.x * 64000;
    for (int i = tid; i < 64000; i += 256) psum[pbase + i] = acc[i];
    for (int i = tid; i < 500;   i += 256) pdeg[(size_t)blockIdx.x * 500 + i] = dg[i];
}

// ---------------- reduce partials -> nbr_mean [NC][128] ----------------
__global__ __launch_bounds__(128) void k_reduce(const float* __restrict__ psum,
    const float* __restrict__ pdeg, float* __restrict__ nbr)
{
    int c = blockIdx.x, f = threadIdx.x;
    int half = c / 500, cl = c - half * 500;
    __shared__ float sdeg;
    if (f == 0) {
        float d = 0.f;
        for (int p = 0; p < 128; ++p) d += pdeg[(size_t)(half * 128 + p) * 500 + cl];
        sdeg = fmaxf(d, 1.f);
    }
    __syncthreads();
    float s = 0.f;
    for (int p = 0; p < 128; ++p)
        s += psum[(size_t)(half * 128 + p) * 64000 + (size_t)cl * 128 + f];
    nbr[(size_t)c * 128 + f] = s / sdeg;
}

// ---------------- per-channel: fus_src = [ch(14) | msg(67) | 0pad(3)] ----------------
__global__ __launch_bounds__(96) void k_channel(const float* __restrict__ chnum,
    const int* __restrict__ chid, const float* __restrict__ cid_emb,
    const float* __restrict__ chan_W, const float* __restrict__ chan_b,
    const float* __restrict__ msg_W, const float* __restrict__ msg_b,
    const float* __restrict__ nbr, float* __restrict__ fus)
{
    int c = blockIdx.x, j = threadIdx.x;
    if (j >= 84) return;
    float val = 0.f;
    if (j < 14) {
        val = chan_b[j];
        for (int k = 0; k < 8; ++k) val += chnum[c * 8 + k] * chan_W[k * 14 + j];
        int id = chid[c];
        for (int k = 0; k < 16; ++k) val += cid_emb[id * 16 + k] * chan_W[(8 + k) * 14 + j];
    } else if (j < 81) {
        int jj = j - 14;
        val = msg_b[jj];
        for (int k = 0; k < 113; ++k) val += nbr[(size_t)c * 128 + k] * msg_W[k * 67 + jj];
    }
    fus[(size_t)c * 84 + j] = val;
}

// ---------------- generic f32 WMMA GEMM: out = act(gather(A) @ W + b) ----------------
// wave = 16-row M tile; W^T staged in LDS; A per-lane float2 matches 16x4 A layout.
template<int NT, int KS, bool RELU, bool HASIDX>
__global__ __launch_bounds__(256) void k_gemm(const float* __restrict__ A, int strideA,
    const int* __restrict__ idx, const float* __restrict__ W,
    const float* __restrict__ bias, int K, int N,
    float* __restrict__ out, int strideO, int colOff, int nStore)
{
    constexpr int KP4 = KS * 4;
    extern __shared__ float smem[];
    float* wt = smem;                 // [NT*16][KP4]  (W transposed, zero padded)
    float* bs = smem + NT * 16 * KP4; // [NT*16]
    const int tid = threadIdx.x;
    for (int i = tid; i < NT * 16 * KP4; i += 256) {
        int n = i / KP4, k = i - n * KP4;
        wt[i] = (n < N && k < K) ? W[k * N + n] : 0.f;
    }
    for (int i = tid; i < NT * 16; i += 256) bs[i] = (i < N) ? bias[i] : 0.f;
    __syncthreads();

    const int wave = tid >> 5, l = tid & 31;
    const int ml = l & 15, hi = l >> 4;
    const int mbase = blockIdx.x * 128 + wave * 16;
    int row = mbase + ml;
    int rsrc = row;
    if (HASIDX) rsrc = idx[row];
    const float* aptr = A + (size_t)rsrc * strideA + (hi << 1);

    v8f c[NT];
    #pragma unroll
    for (int nt = 0; nt < NT; ++nt) {
        v8f zz = {0.f, 0.f, 0.f, 0.f, 0.f, 0.f, 0.f, 0.f};
        c[nt] = zz;
    }

    for (int kk = 0; kk < KS; ++kk) {
        v2f a = *reinterpret_cast<const v2f*>(aptr + kk * 4);
        #pragma unroll
        for (int nt = 0; nt < NT; ++nt) {
            v2f b = *reinterpret_cast<const v2f*>(&wt[(nt * 16 + ml) * KP4 + kk * 4 + (hi << 1)]);
            c[nt] = __builtin_amdgcn_wmma_f32_16x16x4_f32(
                false, a, false, b, (short)0, c[nt], false, false);
        }
    }

    #pragma unroll
    for (int nt = 0; nt < NT; ++nt) {
        int n = nt * 16 + ml;
        float bval = bs[n];
        if (n < nStore) {
            #pragma unroll
            for (int v = 0; v < 8; ++v) {
                float val = c[nt][v] + bval;
                if (RELU) val = fmaxf(val, 0.f);
                int r = mbase + v + (hi << 3);
                out[(size_t)r * strideO + colOff + n] = val;
            }
        }
    }
}

extern "C" void kernel_launch(void* const* d_in, const int* in_sizes, int n_in,
                              void* d_out, int out_size, void* d_ws, size_t ws_size,
                              hipStream_t stream)
{
    (void)in_sizes; (void)n_in; (void)out_size; (void)ws_size;
    const float*    device_num  = (const float*)d_in[0];
    const int*      device_cat  = (const int*)d_in[1];
    const float*    channel_num = (const float*)d_in[2];
    const int*      channel_id  = (const int*)d_in[3];
    const uint32_t* adj32       = (const uint32_t*)d_in[4];   // bool bytes, read as dwords
    const int*      chan_idx    = (const int*)d_in[5];
    const int*      dev_idx     = (const int*)d_in[6];
    const float* emb0 = (const float*)d_in[7];
    const float* emb1 = (const float*)d_in[8];
    const float* emb2 = (const float*)d_in[9];
    const float* emb3 = (const float*)d_in[10];
    const float* emb4 = (const float*)d_in[11];
    const float* emb5 = (const float*)d_in[12];
    const float* emb6 = (const float*)d_in[13];
    const float* cid_emb = (const float*)d_in[14];
    const float* dev_W1 = (const float*)d_in[15];
    const float* dev_b1 = (const float*)d_in[16];
    const float* dev_W2 = (const float*)d_in[17];
    const float* dev_b2 = (const float*)d_in[18];
    const float* chan_W = (const float*)d_in[19];
    const float* chan_b = (const float*)d_in[20];
    const float* msg_W  = (const float*)d_in[21];
    const float* msg_b  = (const float*)d_in[22];
    const float* fus_W  = (const float*)d_in[23];
    const float* fus_b  = (const float*)d_in[24];
    const float* cc_W1  = (const float*)d_in[25];
    const float* cc_b1  = (const float*)d_in[26];
    const float* cc_W2  = (const float*)d_in[27];
    const float* cc_b2  = (const float*)d_in[28];
    const float* cc_W3  = (const float*)d_in[29];
    const float* cc_b3  = (const float*)d_in[30];

    float* ws = (float*)d_ws;
    float* dev_emb = ws + OFF_DEVEMB;
    float* psum    = ws + OFF_PSUM;
    float* pdeg    = ws + OFF_PDEG;
    float* nbr     = ws + OFF_NBR;
    float* fus     = ws + OFF_FUS;
    float* h       = ws + OFF_H;
    float* z       = ws + OFF_Z;    // aliases psum region (agg finished before use)
    float* z1      = ws + OFF_Z1;
    float* z2      = ws + OFF_Z2;
    float* outp    = (float*)d_out;

    // 1) build padded device embeddings
    k_build<<<ND / 2, 256, 0, stream>>>(device_num, device_cat,
        emb0, emb1, emb2, emb3, emb4, emb5, emb6, dev_emb);

    // 2) sparse neighbor aggregation (256KB+ LDS accumulators)
    hipFuncSetAttribute((const void*)k_agg,
        hipFuncAttributeMaxDynamicSharedMemorySize, 262144);
    k_agg<<<256, 256, 262144, stream>>>(adj32, dev_emb, psum, pdeg);

    // 3) reduce partials -> nbr_mean
    k_reduce<<<NC, 128, 0, stream>>>(psum, pdeg, nbr);

    // 4) per-channel fusion source [ch | msg]
    k_channel<<<NC, 96, 0, stream>>>(channel_num, channel_id, cid_emb,
        chan_W, chan_b, msg_W, msg_b, nbr, fus);

    // 5) batched MLP chain, all f32 WMMA 16x16x4
    size_t sh;
    sh = (size_t)(5 * 16 * 116 + 80) * 4;
    k_gemm<5, 29, true, true><<<512, 256, sh, stream>>>(
        dev_emb, 128, dev_idx, dev_W1, dev_b1, 113, 67, h, 80, 0, 80);
    sh = (size_t)(4 * 16 * 68 + 64) * 4;
    k_gemm<4, 17, true, false><<<512, 256, sh, stream>>>(
        h, 80, nullptr, dev_W2, dev_b2, 67, 50, z, 112, 48, 64);
    sh = (size_t)(3 * 16 * 84 + 48) * 4;
    k_gemm<3, 21, true, true><<<512, 256, sh, stream>>>(
        fus, 84, chan_idx, fus_W, fus_b, 81, 48, z, 112, 0, 48);
    sh = (size_t)(4 * 16 * 100 + 64) * 4;
    k_gemm<4, 25, true, false><<<512, 256, sh, stream>>>(
        z, 112, nullptr, cc_W1, cc_b1, 98, 58, z1, 64, 0, 64);
    sh = (size_t)(2 * 16 * 60 + 32) * 4;
    k_gemm<2, 15, true, false><<<512, 256, sh, stream>>>(
        z1, 64, nullptr, cc_W2, cc_b2, 58, 29, z2, 32, 0, 32);
    sh = (size_t)(1 * 16 * 32 + 16) * 4;
    k_gemm<1, 8, false, false><<<512, 256, sh, stream>>>(
        z2, 32, nullptr, cc_W3, cc_b3, 29, 1, outp, 1, 0, 1);
}